// GGNNModel_62912680952417
// MI455X (gfx1250) — compile-verified
//
#include <hip/hip_runtime.h>
#include <hip/hip_bf16.h>

// ---------------------------------------------------------------------------
// GGNN on gfx1250 (MI455X).
// All dense contractions (edge-gate MLP 257x32, GRU 256x128 x3, head 256x128)
// run on V_WMMA_F32_16X16X4_F32 (fp32-exact matrix path). h (25.6 MB) is
// L2-resident so edge gathers are L2 traffic, not HBM. Weights are staged
// pair-interleaved ({row k, row k+1} as float2) so every B-fragment is a
// single b64 load straight into an even-aligned VGPR pair. K-loops are split
// at the concat boundary (K=128) so A-base pointers are loop-invariant (no
// per-step 64-bit cndmask address selects).
// ---------------------------------------------------------------------------

#define NN   50000
#define NE   800000
#define NEP  850000      // NE + NN self loops; 850000 = 16*53125 exactly
#define FIN  16
#define PREC 32
#define CH   128
#define GHD  32
#define NLAYERS 5
#define NTILES_E (NEP/16)   // 53125
#define NTILES_N (NN/16)    // 3125
#define PAIRS   (256/2)     // 128 K-pairs for the 256-deep GEMMs

typedef __attribute__((ext_vector_type(2))) float v2f;
typedef __attribute__((ext_vector_type(8))) float v8f;

__device__ __forceinline__ float sigmoidf_(float x) {
  return 1.0f / (1.0f + __expf(-x));
}

__device__ __forceinline__ v8f wmma_f32_k4(v2f a, v2f b, v8f c) {
  return __builtin_amdgcn_wmma_f32_16x16x4_f32(false, a, false, b,
                                               (short)0, c, false, false);
}

__device__ __forceinline__ v2f ldv2(const float* p) {
  float2 t = *(const float2*)p;
  v2f r; r.x = t.x; r.y = t.y;
  return r;
}

// --------------------------- small utility kernels --------------------------

__global__ void k_fill1(float* deg) {
  int i = blockIdx.x * blockDim.x + threadIdx.x;
  if (i < NN) deg[i] = 1.0f;            // self loop contributes 1 to in-degree
}

__global__ void k_degadd(const int* ei, float* deg) {
  int e = blockIdx.x * blockDim.x + threadIdx.x;
  if (e < NE) atomicAdd(&deg[ei[NE + e]], 1.0f);   // col = ei[1][e]
}

__global__ void k_rsqrt(float* deg) {
  int i = blockIdx.x * blockDim.x + threadIdx.x;
  if (i < NN) deg[i] = rsqrtf(deg[i]);  // deg >= 1 always
}

__global__ void k_zero(float* p, long long n) {
  long long i = (long long)blockIdx.x * blockDim.x + threadIdx.x;
  if (i < n) p[i] = 0.0f;
}

// Pair-interleave a 256x128 weight: out[(k/2)*128 + c] = {W[k][c], W[k+1][c]}
__global__ void k_pairw(const float* __restrict__ W, float2* __restrict__ out) {
  int i = blockIdx.x * blockDim.x + threadIdx.x;
  if (i >= PAIRS * CH) return;
  int p = i / CH, c = i - p * CH;
  float2 t;
  t.x = W[(2 * p) * CH + c];
  t.y = W[(2 * p + 1) * CH + c];
  out[i] = t;
}

// preproc: h0 = relu(relu(x@W_pre+b_pre)@W_init+b_init); ~0.46 GFLOP total.
__global__ void k_pre(const float* __restrict__ x,
                      const float* __restrict__ Wp, const float* __restrict__ bp,
                      const float* __restrict__ Wi, const float* __restrict__ bi,
                      float* __restrict__ h0, float* __restrict__ hA) {
  int nidx = blockIdx.x * blockDim.x + threadIdx.x;
  if (nidx >= NN) return;
  float xin[FIN];
#pragma unroll
  for (int k = 0; k < FIN; k++) xin[k] = x[nidx * FIN + k];
  float pre[PREC];
#pragma unroll
  for (int j = 0; j < PREC; j++) {
    float s = bp[j];
#pragma unroll
    for (int k = 0; k < FIN; k++) s += xin[k] * Wp[k * PREC + j];
    pre[j] = fmaxf(s, 0.0f);
  }
  for (int c = 0; c < CH; c++) {
    float s = bi[c];
#pragma unroll
    for (int k = 0; k < PREC; k++) s += pre[k] * Wi[k * CH + c];
    s = fmaxf(s, 0.0f);
    h0[(size_t)nidx * CH + c] = s;
    hA[(size_t)nidx * CH + c] = s;
  }
}

// ------------------------- edge gate + scatter (WMMA) ------------------------
// One wave per 16-edge tile. G = [x_i | x_j | ea] (16 x 257).
// H = relu(G @ W_g1 + b_g1) via 64 K=4 WMMA steps per 16-col N-tile (2 tiles),
// then gate = sigmoid(H @ W_g2 + b_g2) via shfl_xor reduction,
// then msg = gate*norm*x_j scattered with global_atomic_add_f32.
__global__ void __launch_bounds__(256)
k_edge(const int* __restrict__ ei, const float* __restrict__ eattr,
       const float* __restrict__ dis, const float* __restrict__ h,
       const float* __restrict__ Wg1, const float* __restrict__ bg1,
       const float* __restrict__ Wg2, const float* __restrict__ bg2,
       float* __restrict__ aggr) {
  // pair-interleaved W_g1 rows 0..255: sB[(k/2)*32 + col] = {W[k][col], W[k+1][col]}
  __shared__ float2 sB[PAIRS * GHD];           // 32 KB
  __shared__ float  sW256[GHD];                // row 256 (ea row)
  int tid = threadIdx.x;
  for (int i = tid; i < PAIRS * GHD; i += 256) {
    int p = i >> 5;                            // pair index
    int c = i & 31;
    float2 t;
    t.x = Wg1[(2 * p) * GHD + c];
    t.y = Wg1[(2 * p + 1) * GHD + c];
    sB[i] = t;
  }
  if (tid < GHD) sW256[tid] = Wg1[256 * GHD + tid];
  int l = tid & 31;
  int w = tid >> 5;
  int tile = blockIdx.x * 8 + w;
  __syncthreads();
  if (tile >= NTILES_E) return;                // whole-wave uniform exit

  int m = l & 15;                              // edge-in-tile this lane loads A for
  int e = tile * 16 + m;
  int ri, ci; float eav;
  if (e < NE) { ri = ei[e]; ci = ei[NE + e]; eav = eattr[e]; }
  else        { ri = e - NE; ci = e - NE; eav = 0.0f; }
  float nrm = dis[ri] * dis[ci];
  const float* rowI = h + (size_t)ci * CH;     // x_i = h[col]
  const float* rowJ = h + (size_t)ri * CH;     // x_j = h[row]

  int n = l & 15;                              // output column within N-tile
  int kshift = (l >= 16) ? 2 : 0;
  v8f acc0 = {0.f,0.f,0.f,0.f,0.f,0.f,0.f,0.f};
  v8f acc1 = {0.f,0.f,0.f,0.f,0.f,0.f,0.f,0.f};
  // first half: K = 0..127 comes from x_i = h[col]
#pragma unroll 4
  for (int k0 = 0; k0 < CH; k0 += 4) {
    int k = k0 + kshift;
    v2f a = ldv2(rowI + k);
    int pair = k >> 1;
    v2f b0, b1;
    { float2 t = sB[pair * GHD + n];      b0.x = t.x; b0.y = t.y; }
    { float2 t = sB[pair * GHD + 16 + n]; b1.x = t.x; b1.y = t.y; }
    acc0 = wmma_f32_k4(a, b0, acc0);
    acc1 = wmma_f32_k4(a, b1, acc1);
  }
  // second half: K = 128..255 comes from x_j = h[row]
#pragma unroll 4
  for (int k0 = 0; k0 < CH; k0 += 4) {
    int k = k0 + kshift;
    v2f a = ldv2(rowJ + k);
    int pair = (k + CH) >> 1;
    v2f b0, b1;
    { float2 t = sB[pair * GHD + n];      b0.x = t.x; b0.y = t.y; }
    { float2 t = sB[pair * GHD + 16 + n]; b1.x = t.x; b1.y = t.y; }
    acc0 = wmma_f32_k4(a, b0, acc0);
    acc1 = wmma_f32_k4(a, b1, acc1);
  }

  int hilo = (l >= 16) ? 8 : 0;
  float w256a = sW256[n];
  float w256b = sW256[16 + n];
  float bg1a = bg1[n],  bg1b = bg1[16 + n];
  float wg2a = Wg2[n],  wg2b = Wg2[16 + n];
  float bg2v = bg2[0];
  float pg[8];
#pragma unroll
  for (int r = 0; r < 8; r++) {
    // ea row (K=256) contribution; ea of edge M=r+hilo lives in within-16 lane r+hilo
    float eaM = __shfl(eav, r + hilo, 16);
    float h0v = fmaxf(acc0[r] + eaM * w256a + bg1a, 0.0f);
    float h1v = fmaxf(acc1[r] + eaM * w256b + bg1b, 0.0f);
    float p = h0v * wg2a + h1v * wg2b;         // partial over this lane's 2 cols
    p += __shfl_xor(p, 1, 32);
    p += __shfl_xor(p, 2, 32);
    p += __shfl_xor(p, 4, 32);
    p += __shfl_xor(p, 8, 32);                 // sum over 16 cols in each half
    pg[r] = sigmoidf_(p + bg2v);               // gate for edge M=r+hilo
  }

  // Scatter: 2 lanes per edge, 64 channels each. Lane metadata via shfl.
  int msub = l & 7;
  int mloc = msub + hilo;
  int rm = __shfl(ri, mloc, 16);
  int cm = __shfl(ci, mloc, 16);
  float nm = __shfl(nrm, mloc, 16);
  float g = pg[msub] * nm;                     // gate * norm
  const float* hrow = h + (size_t)rm * CH;
  float* arow = aggr + (size_t)cm * CH;
  int c0 = ((l >> 3) & 1) * 64;
#pragma unroll 4
  for (int j = 0; j < 64; j++) {
    int c = c0 + j;
    atomicAdd(&arow[c], g * hrow[c]);
  }
}

// ----------------------------- GRU z/r (WMMA) -------------------------------
// Block = one 16-node tile, wave w = 16-col tile. Both z and r accumulated in
// one K loop so A-fragments ([h | aggr]) are loaded once. B from the
// pair-interleaved weight copies (1 b64 load per matrix per K-step).
__global__ void __launch_bounds__(256)
k_gru_zr(const float* __restrict__ h, const float* __restrict__ aggr,
         const float2* __restrict__ Wzp, const float* __restrict__ bz,
         const float2* __restrict__ Wrp, const float* __restrict__ br,
         float* __restrict__ zbuf, float* __restrict__ rhbuf) {
  int l = threadIdx.x & 31, w = threadIdx.x >> 5;
  int nb = blockIdx.x * 16, cb = w * 16;
  int m = l & 15, n = l & 15;
  int kshift = (l >= 16) ? 2 : 0;
  const float* rowH = h    + (size_t)(nb + m) * CH;
  const float* rowA = aggr + (size_t)(nb + m) * CH;
  v8f accZ = {0.f,0.f,0.f,0.f,0.f,0.f,0.f,0.f};
  v8f accR = {0.f,0.f,0.f,0.f,0.f,0.f,0.f,0.f};
#pragma unroll 4
  for (int k0 = 0; k0 < CH; k0 += 4) {        // K = 0..127 from h
    int k = k0 + kshift;
    v2f a = ldv2(rowH + k);
    int pi = (k >> 1) * CH + cb + n;
    v2f bzv, brv;
    { float2 t = Wzp[pi]; bzv.x = t.x; bzv.y = t.y; }
    { float2 t = Wrp[pi]; brv.x = t.x; brv.y = t.y; }
    accZ = wmma_f32_k4(a, bzv, accZ);
    accR = wmma_f32_k4(a, brv, accR);
  }
#pragma unroll 4
  for (int k0 = 0; k0 < CH; k0 += 4) {        // K = 128..255 from aggr
    int k = k0 + kshift;
    v2f a = ldv2(rowA + k);
    int pi = ((k + CH) >> 1) * CH + cb + n;
    v2f bzv, brv;
    { float2 t = Wzp[pi]; bzv.x = t.x; bzv.y = t.y; }
    { float2 t = Wrp[pi]; brv.x = t.x; brv.y = t.y; }
    accZ = wmma_f32_k4(a, bzv, accZ);
    accR = wmma_f32_k4(a, brv, accR);
  }
  int hilo = (l >= 16) ? 8 : 0;
  int col = cb + n;
  float bzc = bz[col], brc = br[col];
#pragma unroll
  for (int r = 0; r < 8; r++) {
    int node = nb + r + hilo;
    size_t idx = (size_t)node * CH + col;
    float z  = sigmoidf_(accZ[r] + bzc);
    float rr = sigmoidf_(accR[r] + brc);
    zbuf[idx]  = z;
    rhbuf[idx] = rr * h[idx];
  }
}

// --------------------- GRU candidate + combine (WMMA) -----------------------
__global__ void __launch_bounds__(256)
k_gru_h(const float* __restrict__ h, const float* __restrict__ aggr,
        const float* __restrict__ rhbuf, const float* __restrict__ zbuf,
        const float2* __restrict__ Whp, const float* __restrict__ bh,
        float* __restrict__ hout) {
  int l = threadIdx.x & 31, w = threadIdx.x >> 5;
  int nb = blockIdx.x * 16, cb = w * 16;
  int m = l & 15, n = l & 15;
  int kshift = (l >= 16) ? 2 : 0;
  const float* row0 = rhbuf + (size_t)(nb + m) * CH;   // r*h
  const float* row1 = aggr  + (size_t)(nb + m) * CH;
  v8f acc = {0.f,0.f,0.f,0.f,0.f,0.f,0.f,0.f};
#pragma unroll 4
  for (int k0 = 0; k0 < CH; k0 += 4) {        // K = 0..127 from r*h
    int k = k0 + kshift;
    v2f a = ldv2(row0 + k);
    v2f b;
    { float2 t = Whp[(k >> 1) * CH + cb + n]; b.x = t.x; b.y = t.y; }
    acc = wmma_f32_k4(a, b, acc);
  }
#pragma unroll 4
  for (int k0 = 0; k0 < CH; k0 += 4) {        // K = 128..255 from aggr
    int k = k0 + kshift;
    v2f a = ldv2(row1 + k);
    v2f b;
    { float2 t = Whp[((k + CH) >> 1) * CH + cb + n]; b.x = t.x; b.y = t.y; }
    acc = wmma_f32_k4(a, b, acc);
  }
  int hilo = (l >= 16) ? 8 : 0;
  int col = cb + n;
  float bhc = bh[col];
#pragma unroll
  for (int r = 0; r < 8; r++) {
    int node = nb + r + hilo;
    size_t idx = (size_t)node * CH + col;
    float hc = fmaxf(acc[r] + bhc, 0.0f);
    float z = zbuf[idx];
    hout[idx] = (1.0f - z) * h[idx] + z * hc;
  }
}

// ------------------------------ head GEMM (WMMA) ----------------------------
__global__ void __launch_bounds__(256)
k_head1(const float* __restrict__ h0, const float* __restrict__ hfin,
        const float2* __restrict__ Wf1p, const float* __restrict__ bf1,
        float* __restrict__ obuf) {
  int l = threadIdx.x & 31, w = threadIdx.x >> 5;
  int nb = blockIdx.x * 16, cb = w * 16;
  int m = l & 15, n = l & 15;
  int kshift = (l >= 16) ? 2 : 0;
  const float* row0 = h0   + (size_t)(nb + m) * CH;
  const float* row1 = hfin + (size_t)(nb + m) * CH;
  v8f acc = {0.f,0.f,0.f,0.f,0.f,0.f,0.f,0.f};
#pragma unroll 4
  for (int k0 = 0; k0 < CH; k0 += 4) {        // K = 0..127 from h0
    int k = k0 + kshift;
    v2f a = ldv2(row0 + k);
    v2f b;
    { float2 t = Wf1p[(k >> 1) * CH + cb + n]; b.x = t.x; b.y = t.y; }
    acc = wmma_f32_k4(a, b, acc);
  }
#pragma unroll 4
  for (int k0 = 0; k0 < CH; k0 += 4) {        // K = 128..255 from final h
    int k = k0 + kshift;
    v2f a = ldv2(row1 + k);
    v2f b;
    { float2 t = Wf1p[((k + CH) >> 1) * CH + cb + n]; b.x = t.x; b.y = t.y; }
    acc = wmma_f32_k4(a, b, acc);
  }
  int hilo = (l >= 16) ? 8 : 0;
  int col = cb + n;
  float bc = bf1[col];
#pragma unroll
  for (int r = 0; r < 8; r++) {
    int node = nb + r + hilo;
    obuf[(size_t)node * CH + col] = fmaxf(acc[r] + bc, 0.0f);
  }
}

__global__ void k_head2(const float* __restrict__ obuf,
                        const float* __restrict__ Wf2, const float* __restrict__ bf2,
                        float* __restrict__ out) {
  int nidx = blockIdx.x * blockDim.x + threadIdx.x;
  if (nidx >= NN) return;
  const float* row = obuf + (size_t)nidx * CH;
  float s0 = bf2[0], s1 = bf2[1];
#pragma unroll 4
  for (int c = 0; c < CH; c++) {
    float v = row[c];
    s0 += v * Wf2[c * 2 + 0];
    s1 += v * Wf2[c * 2 + 1];
  }
  out[nidx * 2 + 0] = s0;
  out[nidx * 2 + 1] = s1;
}

// --------------------------------- launch -----------------------------------

extern "C" void kernel_launch(void* const* d_in, const int* in_sizes, int n_in,
                              void* d_out, int out_size, void* d_ws, size_t ws_size,
                              hipStream_t stream) {
  const float* x     = (const float*)d_in[0];
  const int*   ei    = (const int*)  d_in[1];
  const float* ea    = (const float*)d_in[2];
  const float* Wpre  = (const float*)d_in[3];
  const float* bpre  = (const float*)d_in[4];
  const float* Winit = (const float*)d_in[5];
  const float* binit = (const float*)d_in[6];
  const float* Wg1   = (const float*)d_in[7];
  const float* bg1   = (const float*)d_in[8];
  const float* Wg2   = (const float*)d_in[9];
  const float* bg2   = (const float*)d_in[10];
  const float* Wz    = (const float*)d_in[11];
  const float* bz    = (const float*)d_in[12];
  const float* Wr    = (const float*)d_in[13];
  const float* br    = (const float*)d_in[14];
  const float* Wh    = (const float*)d_in[15];
  const float* bh    = (const float*)d_in[16];
  const float* Wf1   = (const float*)d_in[17];
  const float* bf1   = (const float*)d_in[18];
  const float* Wf2   = (const float*)d_in[19];
  const float* bf2   = (const float*)d_in[20];
  float* out = (float*)d_out;

  // workspace: dis[N], h0, hA, hB, aggr, zbuf, rhbuf (N*C each),
  // then 4 pair-interleaved weight copies (PAIRS*CH float2 each)
  float* ws    = (float*)d_ws;
  float* dis   = ws;
  float* h0    = dis  + NN;
  float* hA    = h0   + (size_t)NN * CH;
  float* hB    = hA   + (size_t)NN * CH;
  float* aggr  = hB   + (size_t)NN * CH;
  float* zbuf  = aggr + (size_t)NN * CH;
  float* rhbuf = zbuf + (size_t)NN * CH;
  float2* Wzp  = (float2*)(rhbuf + (size_t)NN * CH);
  float2* Wrp  = Wzp + PAIRS * CH;
  float2* Whp  = Wrp + PAIRS * CH;
  float2* Wf1p = Whp + PAIRS * CH;

  dim3 blk(256);
  int pw_blocks = (PAIRS * CH + 255) / 256;
  k_pairw <<<pw_blocks, blk, 0, stream>>>(Wz,  Wzp);
  k_pairw <<<pw_blocks, blk, 0, stream>>>(Wr,  Wrp);
  k_pairw <<<pw_blocks, blk, 0, stream>>>(Wh,  Whp);
  k_pairw <<<pw_blocks, blk, 0, stream>>>(Wf1, Wf1p);

  k_fill1 <<<(NN + 255) / 256, blk, 0, stream>>>(dis);
  k_degadd<<<(NE + 255) / 256, blk, 0, stream>>>(ei, dis);
  k_rsqrt <<<(NN + 255) / 256, blk, 0, stream>>>(dis);
  k_pre   <<<(NN + 255) / 256, blk, 0, stream>>>(x, Wpre, bpre, Winit, binit, h0, hA);

  float* hc = hA;
  float* hn = hB;
  for (int layer = 0; layer < NLAYERS; layer++) {
    long long nc = (long long)NN * CH;
    k_zero  <<<(unsigned)((nc + 255) / 256), blk, 0, stream>>>(aggr, nc);
    k_edge  <<<(NTILES_E + 7) / 8, blk, 0, stream>>>(ei, ea, dis, hc, Wg1, bg1, Wg2, bg2, aggr);
    k_gru_zr<<<NTILES_N, blk, 0, stream>>>(hc, aggr, Wzp, bz, Wrp, br, zbuf, rhbuf);
    k_gru_h <<<NTILES_N, blk, 0, stream>>>(hc, aggr, rhbuf, zbuf, Whp, bh, hn);
    float* t = hc; hc = hn; hn = t;
  }

  k_head1<<<NTILES_N, blk, 0, stream>>>(h0, hc, Wf1p, bf1, rhbuf);
  k_head2<<<(NN + 255) / 256, blk, 0, stream>>>(rhbuf, Wf2, bf2, out);
}